// TransitionDown_37091337568770
// MI455X (gfx1250) — compile-verified
//
#include <hip/hip_runtime.h>

#define B_    4
#define N_    8192
#define S_    2048      // NPOINT
#define K_    16
#define INDIM 64
#define CIN   67        // 3 + INDIM
#define OUT_  128
#define EPS_  1e-5f

typedef __attribute__((ext_vector_type(2))) float v2f;
typedef __attribute__((ext_vector_type(8))) float v8f;

// ---------------------------------------------------------------------------
// Kernel 1: Farthest point sampling. One block per batch, 256 threads,
// 32 points per thread kept as register-resident running min-distances.
// Argmax per iteration: per-thread scan -> wave32 shfl_xor reduce -> 8-wave
// LDS combine. Tie-break: smallest index (matches jnp.argmax).
// Writes new_xyz (gathered points) directly into d_out.
// ---------------------------------------------------------------------------
#define FPS_T 256
#define FPS_P (N_ / FPS_T)   // 32

__global__ void fps_kernel(const float* __restrict__ xyz,
                           float* __restrict__ out_xyz) {
    const int b   = blockIdx.x;
    const int tid = threadIdx.x;
    const float* X = xyz + (size_t)b * N_ * 3;

    float dist[FPS_P];
#pragma unroll
    for (int j = 0; j < FPS_P; ++j) dist[j] = 1e10f;

    __shared__ float swv[FPS_T / 32];
    __shared__ int   swi[FPS_T / 32];
    __shared__ int   sfar;

    int far = 0;
    for (int t = 0; t < S_; ++t) {
        // output the current farthest index's coordinates (scan emits pre-update `far`)
        if (tid == 0) {
            out_xyz[((size_t)b * S_ + t) * 3 + 0] = X[far * 3 + 0];
            out_xyz[((size_t)b * S_ + t) * 3 + 1] = X[far * 3 + 1];
            out_xyz[((size_t)b * S_ + t) * 3 + 2] = X[far * 3 + 2];
        }
        const float cx = X[far * 3 + 0];
        const float cy = X[far * 3 + 1];
        const float cz = X[far * 3 + 2];

        float bv = -1.0f;
        int   bi = 0;
#pragma unroll
        for (int j = 0; j < FPS_P; ++j) {
            const int p = j * FPS_T + tid;
            const float dx = X[p * 3 + 0] - cx;
            const float dy = X[p * 3 + 1] - cy;
            const float dz = X[p * 3 + 2] - cz;
            const float d  = dx * dx + dy * dy + dz * dz;
            const float nd = fminf(dist[j], d);
            dist[j] = nd;
            if (nd > bv || (nd == bv && p < bi)) { bv = nd; bi = p; }
        }
        // wave32 argmax reduction (min-index tie-break)
#pragma unroll
        for (int m = 16; m >= 1; m >>= 1) {
            const float ov = __shfl_xor(bv, m, 32);
            const int   oi = __shfl_xor(bi, m, 32);
            if (ov > bv || (ov == bv && oi < bi)) { bv = ov; bi = oi; }
        }
        const int wid = tid >> 5;
        if ((tid & 31) == 0) { swv[wid] = bv; swi[wid] = bi; }
        __syncthreads();
        if (tid == 0) {
            float fv = swv[0]; int fi = swi[0];
#pragma unroll
            for (int w = 1; w < FPS_T / 32; ++w)
                if (swv[w] > fv || (swv[w] == fv && swi[w] < fi)) { fv = swv[w]; fi = swi[w]; }
            sfar = fi;
        }
        __syncthreads();
        far = sfar;
    }
}

// ---------------------------------------------------------------------------
// Kernel 2: KNN (K=16) per query. One thread per query; register-resident
// sorted top-16 with fully unrolled insertion (stable on ties, matching
// top_k of the matmul-form squared distance).
// ---------------------------------------------------------------------------
__global__ void knn_kernel(const float* __restrict__ xyz,
                           const float* __restrict__ new_xyz,
                           int* __restrict__ knn) {
    const int q = blockIdx.x * blockDim.x + threadIdx.x;   // 0 .. B*S-1
    const int b = q >> 11;                                 // q / S_
    const float* X = xyz + (size_t)b * N_ * 3;

    const float qx = new_xyz[q * 3 + 0];
    const float qy = new_xyz[q * 3 + 1];
    const float qz = new_xyz[q * 3 + 2];
    const float q2 = qx * qx + qy * qy + qz * qz;

    float bd[K_];
    int   bi[K_];
#pragma unroll
    for (int t = 0; t < K_; ++t) { bd[t] = 3.4e38f; bi[t] = 0; }

    for (int j = 0; j < N_; ++j) {
        const float x = X[j * 3 + 0];
        const float y = X[j * 3 + 1];
        const float z = X[j * 3 + 2];
        const float x2  = x * x + y * y + z * z;
        const float dot = qx * x + qy * y + qz * z;
        const float d   = q2 + x2 - 2.0f * dot;
        if (d < bd[K_ - 1]) {
            bd[K_ - 1] = d; bi[K_ - 1] = j;
#pragma unroll
            for (int t = K_ - 1; t > 0; --t) {
                if (bd[t] < bd[t - 1]) {
                    const float tv = bd[t]; bd[t] = bd[t - 1]; bd[t - 1] = tv;
                    const int   ti = bi[t]; bi[t] = bi[t - 1]; bi[t - 1] = ti;
                }
            }
        }
    }
#pragma unroll
    for (int t = 0; t < K_; ++t) knn[q * K_ + t] = bi[t];
}

// ---------------------------------------------------------------------------
// Kernel 3: fused gather -> GEMM1(16x68 @ 68x128, relu) -> GEMM2(16x128 @
// 128x128) -> max over K rows -> LayerNorm. One wave32 per query, WMMA
// V_WMMA_F32_16X16X4_F32 for both GEMMs. 4 waves / 128-thread block.
// LDS strides 68 and 132 (both == 4 mod 64) -> bank-conflict-free fragment
// loads. Weights stream from global (L2 resident, reused by all 8192 waves).
// ---------------------------------------------------------------------------
#define WAVES 4

__global__ void mlp_kernel(const float* __restrict__ xyz,
                           const float* __restrict__ feat,
                           const float* __restrict__ new_xyz,
                           const int*   __restrict__ knn,
                           const float* __restrict__ W1,
                           const float* __restrict__ b1,
                           const float* __restrict__ W2,
                           const float* __restrict__ b2,
                           const float* __restrict__ lng,
                           const float* __restrict__ lnb,
                           float* __restrict__ out_feat) {
    __shared__ float Ag[WAVES][16 * 68];    // grouped tile, K-padded (col 67 = 0)
    __shared__ float Hs[WAVES][16 * 132];   // relu(h1), padded stride

    const int lane = threadIdx.x & 31;
    const int wid  = threadIdx.x >> 5;
    const int q    = blockIdx.x * WAVES + wid;    // grid covers exactly B*S
    const int b    = q >> 11;                     // q / S_

    float* A    = Ag[wid];
    float* Hrow = Hs[wid];

    // ---- gather grouped = [xyz(nbr)-center | feat(nbr)] into LDS -----------
    const float cx = new_xyz[q * 3 + 0];
    const float cy = new_xyz[q * 3 + 1];
    const float cz = new_xyz[q * 3 + 2];
    const int nb = knn[q * K_ + (lane & 15)];
    for (int r = 0; r < K_; ++r) {
        const int nr = __shfl(nb, r, 32);
        const float* px = xyz  + ((size_t)b * N_ + nr) * 3;
        const float* pf = feat + ((size_t)b * N_ + nr) * INDIM;
        for (int c = lane; c < 68; c += 32) {
            float v;
            if (c == 67)      v = 0.0f;
            else if (c >= 3)  v = pf[c - 3];
            else              v = px[c] - ((c == 0) ? cx : ((c == 1) ? cy : cz));
            A[r * 68 + c] = v;
        }
    }

    const int row   = lane & 15;
    const int koff  = (lane < 16) ? 0 : 2;   // A/B fragment K-offset per half-wave
    const int mbase = (lane < 16) ? 0 : 8;   // C/D row base per half-wave
    float pooled[8];

    // ---- GEMM1: h1 = relu(grouped @ W1 + b1), 17 K-tiles of 4 ---------------
    for (int n = 0; n < 8; ++n) {
        const int colN = n * 16 + row;
        v8f acc = {0.f, 0.f, 0.f, 0.f, 0.f, 0.f, 0.f, 0.f};
        for (int t = 0; t < 17; ++t) {
            const int k0 = 4 * t + koff;
            v2f av = { A[row * 68 + k0], A[row * 68 + k0 + 1] };
            const float w0 = (k0     < CIN) ? W1[(size_t)k0 * OUT_ + colN]       : 0.f;
            const float w1 = (k0 + 1 < CIN) ? W1[(size_t)(k0 + 1) * OUT_ + colN] : 0.f;
            v2f bv = { w0, w1 };
            acc = __builtin_amdgcn_wmma_f32_16x16x4_f32(
                false, av, false, bv, (short)0, acc, false, false);
        }
        const float bias = b1[colN];
#pragma unroll
        for (int r = 0; r < 8; ++r) {
            const float h = acc[r] + bias;
            Hrow[(mbase + r) * 132 + colN] = fmaxf(h, 0.0f);
        }
    }

    // ---- GEMM2: h2 = h1 @ W2 + b2, then max over the 16 neighbor rows -------
    for (int n = 0; n < 8; ++n) {
        const int colN = n * 16 + row;
        v8f acc = {0.f, 0.f, 0.f, 0.f, 0.f, 0.f, 0.f, 0.f};
        for (int t = 0; t < 32; ++t) {
            const int k0 = 4 * t + koff;
            v2f av = { Hrow[row * 132 + k0], Hrow[row * 132 + k0 + 1] };
            v2f bv = { W2[(size_t)k0 * OUT_ + colN], W2[(size_t)(k0 + 1) * OUT_ + colN] };
            acc = __builtin_amdgcn_wmma_f32_16x16x4_f32(
                false, av, false, bv, (short)0, acc, false, false);
        }
        const float bias = b2[colN];
        float m = -3.4e38f;
#pragma unroll
        for (int r = 0; r < 8; ++r) m = fmaxf(m, acc[r] + bias);
        const float mo = __shfl_xor(m, 16, 32);   // combine the two 8-row halves
        pooled[n] = fmaxf(m, mo);                 // column max over all 16 rows
    }

    // ---- LayerNorm over 128 channels (lane<16 holds 8 channels each) --------
    float lsum = 0.0f;
#pragma unroll
    for (int n = 0; n < 8; ++n) lsum += pooled[n];
    if (lane >= 16) lsum = 0.0f;                  // lanes 16-31 are duplicates
#pragma unroll
    for (int m = 16; m >= 1; m >>= 1) lsum += __shfl_xor(lsum, m, 32);
    const float mu = lsum * (1.0f / 128.0f);

    float lvar = 0.0f;
#pragma unroll
    for (int n = 0; n < 8; ++n) { const float d = pooled[n] - mu; lvar += d * d; }
    if (lane >= 16) lvar = 0.0f;
#pragma unroll
    for (int m = 16; m >= 1; m >>= 1) lvar += __shfl_xor(lvar, m, 32);
    const float rs = rsqrtf(lvar * (1.0f / 128.0f) + EPS_);

    if (lane < 16) {
#pragma unroll
        for (int n = 0; n < 8; ++n) {
            const int c = n * 16 + lane;
            out_feat[(size_t)q * OUT_ + c] = (pooled[n] - mu) * rs * lng[c] + lnb[c];
        }
    }
}

// ---------------------------------------------------------------------------
extern "C" void kernel_launch(void* const* d_in, const int* in_sizes, int n_in,
                              void* d_out, int out_size, void* d_ws, size_t ws_size,
                              hipStream_t stream) {
    (void)in_sizes; (void)n_in; (void)out_size; (void)ws_size;
    const float* xyz  = (const float*)d_in[0];
    const float* feat = (const float*)d_in[1];
    const float* W1   = (const float*)d_in[2];
    const float* b1   = (const float*)d_in[3];
    const float* W2   = (const float*)d_in[4];
    const float* b2   = (const float*)d_in[5];
    const float* lng  = (const float*)d_in[6];
    const float* lnb  = (const float*)d_in[7];

    float* out      = (float*)d_out;
    float* out_xyz  = out;                          // (B,S,3)
    float* out_feat = out + (size_t)B_ * S_ * 3;    // (B,S,128)

    int* knn = (int*)d_ws;                          // (B,S,K) = 512 KB

    fps_kernel<<<B_, FPS_T, 0, stream>>>(xyz, out_xyz);
    knn_kernel<<<(B_ * S_) / 256, 256, 0, stream>>>(xyz, out_xyz, knn);
    mlp_kernel<<<(B_ * S_) / WAVES, WAVES * 32, 0, stream>>>(
        xyz, feat, out_xyz, knn, W1, b1, W2, b2, lng, lnb, out_feat);
}